// LidarCameraFusionMambaBlock_16449724745540
// MI455X (gfx1250) — compile-verified
//
#include <hip/hip_runtime.h>
#include <cmath>

typedef __attribute__((ext_vector_type(2))) float v2f;
typedef __attribute__((ext_vector_type(8))) float v8f;

#define NTOK    16384   // 4 * 4096
#define SEQLEN  4096
#define BATCH   4
#define DIN     512
#define DSTATE  16

#if __has_builtin(__builtin_amdgcn_wmma_f32_16x16x4_f32)
#define USE_WMMA_F32 1
#endif

__device__ __forceinline__ v8f mma16x16x4(v2f a, v2f b, v8f c) {
#ifdef USE_WMMA_F32
    // (neg_a, A, neg_b, B, c_mod, C, reuse_a, reuse_b)
    return __builtin_amdgcn_wmma_f32_16x16x4_f32(false, a, false, b, (short)0, c,
                                                 false, false);
#else
    c[0] += a[0]*b[0] + a[1]*b[1];
    return c;
#endif
}

// ---------------------------------------------------------------------------
// Generic GEMM: out[M,N] = X[M,K] @ W[N,K]^T  (+ fused epilogue)
// EP = 0 : plain store
// EP = 1 : out = extra[m,n] * sigmoid(acc + bias[n])     (guidance gate)
// EP = 2 : out = softplus(acc + bias[n])                 (delta)
// GUARD  : column-group bounds check (wave-uniform; only needed when N%64!=0)
// Wave tile: 32(M) x 64(N). Block = 256 threads = 8 waves stacked along M.
// Grid: (M/256, ceil(N/64)). M must be a multiple of 256 (16384 is).
// ---------------------------------------------------------------------------
template<int EP, bool GUARD>
__global__ __launch_bounds__(256)
void gemm_xwt(const float* __restrict__ X, int ldx,
              const float* __restrict__ W, int ldw,
              const float* __restrict__ bias,
              const float* __restrict__ extra, int ldextra,
              float* __restrict__ out, int ldo,
              int N, int K)
{
    const int lane  = threadIdx.x & 31;
    const int wave  = threadIdx.x >> 5;
    const int m0    = blockIdx.x * 256 + wave * 32;
    const int n0    = blockIdx.y * 64;
    const int mrow  = lane & 15;     // row within A frag / col within B,C frag
    const int khalf = lane >> 4;     // 0: K{0,1}, 1: K{2,3}

    v8f acc[2][4];
    #pragma unroll
    for (int i = 0; i < 2; ++i)
        #pragma unroll
        for (int j = 0; j < 4; ++j)
            acc[i][j] = (v8f)(0.0f);

    const float* xr0 = X + (size_t)(m0 + mrow)      * ldx + 2*khalf;
    const float* xr1 = X + (size_t)(m0 + 16 + mrow) * ldx + 2*khalf;

    // Wave-uniform per-group validity (no lane term -> scalar branch, no EXEC churn)
    bool gok[4];
    const float* wr[4];
    #pragma unroll
    for (int j = 0; j < 4; ++j) {
        int gn = n0 + j*16;
        gok[j] = GUARD ? (gn < N) : true;
        wr[j]  = W + (size_t)((GUARD && !gok[j]) ? mrow : (gn + mrow)) * ldw + 2*khalf;
    }

    // ---- software-pipelined K loop: prefetch next frags before using current
    v2f a0 = *(const v2f*)(xr0);
    v2f a1 = *(const v2f*)(xr1);
    v2f bf[4];
    #pragma unroll
    for (int j = 0; j < 4; ++j) {
        bf[j] = (v2f){0.0f, 0.0f};
        if (gok[j]) bf[j] = *(const v2f*)(wr[j]);
    }

    for (int kb = 0; kb < K; kb += 4) {
        const int kn = kb + 4;
        v2f na0 = (v2f){0.0f, 0.0f}, na1 = (v2f){0.0f, 0.0f};
        v2f nbf[4];
        #pragma unroll
        for (int j = 0; j < 4; ++j) nbf[j] = (v2f){0.0f, 0.0f};
        if (kn < K) {                       // uniform scalar branch
            na0 = *(const v2f*)(xr0 + kn);
            na1 = *(const v2f*)(xr1 + kn);
            #pragma unroll
            for (int j = 0; j < 4; ++j)
                if (gok[j]) nbf[j] = *(const v2f*)(wr[j] + kn);
        }
        #pragma unroll
        for (int j = 0; j < 4; ++j) {
            acc[0][j] = mma16x16x4(a0, bf[j], acc[0][j]);
            acc[1][j] = mma16x16x4(a1, bf[j], acc[1][j]);
        }
        a0 = na0; a1 = na1;
        #pragma unroll
        for (int j = 0; j < 4; ++j) bf[j] = nbf[j];
    }

    // D layout: lane holds column n0+j*16+mrow; VGPR v holds row (8*khalf + v)
    #pragma unroll
    for (int mi = 0; mi < 2; ++mi) {
        #pragma unroll
        for (int j = 0; j < 4; ++j) {
            int n = n0 + j*16 + mrow;
            if (!GUARD || gok[j]) {
                #pragma unroll
                for (int v = 0; v < 8; ++v) {
                    int   m   = m0 + mi*16 + khalf*8 + v;
                    float val = acc[mi][j][v];
                    if (EP == 1) {
                        float s = val + bias[n];
                        float g = 1.0f / (1.0f + __expf(-s));
                        val = extra[(size_t)m * ldextra + n] * g;
                    } else if (EP == 2) {
                        float s = val + bias[n];
                        val = (s > 20.0f) ? s : log1pf(__expf(s));
                    }
                    out[(size_t)m * ldo + n] = val;
                }
            }
        }
    }
}

// ---------------------------------------------------------------------------
// Selective scan (one direction). One thread per (b, d) channel.
// grid = (BATCH, DIN/128), block = 128.  ssm rows: [dt(16) | B(16) | C(16)]
// ---------------------------------------------------------------------------
__global__ __launch_bounds__(128)
void scan_kernel(const float* __restrict__ u,      // x_guided [NTOK, DIN]
                 const float* __restrict__ delta,  // [NTOK, DIN]
                 const float* __restrict__ ssm,    // [NTOK, 48]
                 const float* __restrict__ A_log,  // [DIN, DSTATE]
                 float* __restrict__ y,            // [NTOK, DIN]
                 int rev)
{
    const int b = blockIdx.x;
    const int d = blockIdx.y * 128 + threadIdx.x;

    __shared__ float sBC[32];

    float Arow[DSTATE];
    #pragma unroll
    for (int n = 0; n < DSTATE; ++n)
        Arow[n] = -__expf(A_log[d * DSTATE + n]);

    float xs[DSTATE];
    #pragma unroll
    for (int n = 0; n < DSTATE; ++n) xs[n] = 0.0f;

    const size_t base = (size_t)b * SEQLEN;
    for (int t = 0; t < SEQLEN; ++t) {
        const int    tt  = rev ? (SEQLEN - 1 - t) : t;
        const size_t tok = base + tt;

        if (threadIdx.x < 32) sBC[threadIdx.x] = ssm[tok * 48 + 16 + threadIdx.x];
        __syncthreads();

        float dv = delta[tok * DIN + d];
        float uv = u[tok * DIN + d];
        float du = dv * uv;
        float accy = 0.0f;
        #pragma unroll
        for (int n = 0; n < DSTATE; ++n) {
            float dA = __expf(dv * Arow[n]);
            xs[n] = dA * xs[n] + du * sBC[n];
            accy += xs[n] * sBC[16 + n];
        }
        y[tok * DIN + d] = accy;
        __syncthreads();
    }
}

// ---------------------------------------------------------------------------
// y_fused = (y_f + y_b + 2 * u * D[d]) * silu(z),  z = cam_xz[m, 512 + d]
// ---------------------------------------------------------------------------
__global__ __launch_bounds__(256)
void fuse_kernel(const float* __restrict__ yf, const float* __restrict__ yb,
                 const float* __restrict__ u,  const float* __restrict__ cam_xz,
                 const float* __restrict__ Dv, float* __restrict__ yfused)
{
    size_t i = (size_t)blockIdx.x * blockDim.x + threadIdx.x;
    if (i >= (size_t)NTOK * DIN) return;
    int    d = (int)(i & (DIN - 1));
    size_t m = i >> 9;               // / 512
    float  z = cam_xz[m * 1024 + DIN + d];
    float  gate = z / (1.0f + __expf(-z));
    float  skip = u[i] * Dv[d];
    yfused[i] = (yf[i] + yb[i] + 2.0f * skip) * gate;
}

// ---------------------------------------------------------------------------
extern "C" void kernel_launch(void* const* d_in, const int* in_sizes, int n_in,
                              void* d_out, int out_size, void* d_ws, size_t ws_size,
                              hipStream_t stream)
{
    const float* lidar   = (const float*)d_in[0];
    const float* camera  = (const float*)d_in[1];
    const float* W_in    = (const float*)d_in[2];
    const float* W_xproj = (const float*)d_in[3];
    const float* W_dt    = (const float*)d_in[4];
    const float* b_dt    = (const float*)d_in[5];
    const float* A_h2t   = (const float*)d_in[6];
    const float* A_t2h   = (const float*)d_in[7];
    const float* Dvec    = (const float*)d_in[8];
    const float* W_guid  = (const float*)d_in[9];
    const float* b_guid  = (const float*)d_in[10];
    const float* W_out   = (const float*)d_in[11];
    float* out = (float*)d_out;

    // workspace layout (floats); delta / y_fused alias the lid_x buffer
    float* ws       = (float*)d_ws;
    float* cam_xz   = ws;                              // 16384*1024
    float* lid_x    = cam_xz  + (size_t)NTOK * 1024;   // 16384*512 (alias: delta, y_fused)
    float* x_guided = lid_x   + (size_t)NTOK * DIN;    // 16384*512
    float* ssm      = x_guided+ (size_t)NTOK * DIN;    // 16384*48
    float* y_f      = ssm     + (size_t)NTOK * 48;     // 16384*512
    float* y_b      = y_f     + (size_t)NTOK * DIN;    // 16384*512
    float* delta    = lid_x;
    float* y_fused  = lid_x;

    const dim3 blk(256);

    // 1) cam_xz = camera @ W_in^T           [16384,1024]
    gemm_xwt<0,false><<<dim3(NTOK/256, 1024/64), blk, 0, stream>>>(
        camera, 256, W_in, 256, nullptr, nullptr, 0, cam_xz, 1024, 1024, 256);

    // 2) lid_x = lidar @ W_in[:512]^T       [16384,512]
    gemm_xwt<0,false><<<dim3(NTOK/256, DIN/64), blk, 0, stream>>>(
        lidar, 256, W_in, 256, nullptr, nullptr, 0, lid_x, DIN, DIN, 256);

    // 3) x_guided = cam_x * sigmoid(lid_x @ W_guid^T + b_guid)
    gemm_xwt<1,false><<<dim3(NTOK/256, DIN/64), blk, 0, stream>>>(
        lid_x, DIN, W_guid, DIN, b_guid, cam_xz, 1024, x_guided, DIN, DIN, DIN);

    // 4) ssm = cam_x @ W_xproj^T            [16384,48]  (needs column guard)
    gemm_xwt<0,true><<<dim3(NTOK/256, 1), blk, 0, stream>>>(
        cam_xz, 1024, W_xproj, DIN, nullptr, nullptr, 0, ssm, 48, 48, DIN);

    // 5) delta = softplus(dt_low @ W_dt^T + b_dt)   [16384,512]
    gemm_xwt<2,false><<<dim3(NTOK/256, DIN/64), blk, 0, stream>>>(
        ssm, 48, W_dt, 16, b_dt, nullptr, 0, delta, DIN, DIN, 16);

    // 6/7) bidirectional selective scans
    scan_kernel<<<dim3(BATCH, DIN/128), dim3(128), 0, stream>>>(
        x_guided, delta, ssm, A_h2t, y_f, 0);
    scan_kernel<<<dim3(BATCH, DIN/128), dim3(128), 0, stream>>>(
        x_guided, delta, ssm, A_t2h, y_b, 1);

    // 8) y_fused = (y_f + y_b + 2*skip) * silu(cam_z)
    {
        size_t total = (size_t)NTOK * DIN;
        fuse_kernel<<<dim3((unsigned)((total + 255) / 256)), blk, 0, stream>>>(
            y_f, y_b, x_guided, cam_xz, Dvec, y_fused);
    }

    // 9) out = y_fused @ W_out^T            [16384,256]
    gemm_xwt<0,false><<<dim3(NTOK/256, 256/64), blk, 0, stream>>>(
        y_fused, DIN, W_out, DIN, nullptr, nullptr, 0, out, 256, 256, DIN);
}